// Denoise_15427522527245
// MI455X (gfx1250) — compile-verified
//
#include <hip/hip_runtime.h>
#include <hip/hip_bf16.h>
#include <stdint.h>

#define IMG_H 1024
#define IMG_W 1024
#define TILE_W 128
#define TILE_H 64
#define LDS_ROWS (TILE_H + 2)        // 66
#define LDS_STRIDE 136               // cols: 3=left halo, 4..131 interior, 132=right halo
#define NTHREADS 256
#define ROWS_PER_THREAD (TILE_H / 8) // 8  (one wave == one row group)

typedef float        v4f  __attribute__((ext_vector_type(4)));
typedef unsigned int u32x4 __attribute__((ext_vector_type(4)));
typedef int          i32x4 __attribute__((ext_vector_type(4)));
typedef int          i32x8 __attribute__((ext_vector_type(8)));

__device__ __forceinline__ float fmin3(float a, float b, float c) {
    return fminf(fminf(a, b), c);    // -> v_min3_f32
}
__device__ __forceinline__ float fmax3(float a, float b, float c) {
    return fmaxf(fmaxf(a, b), c);    // -> v_max3_f32
}
__device__ __forceinline__ float fmed3(float a, float b, float c) {
    return __builtin_amdgcn_fmed3f(a, b, c);   // v_med3_f32
}

// reflect index for pad==1: valid for i in [-1, n]
__device__ __forceinline__ int reflect1(int i, int n) {
    i = (i < 0) ? -i : i;
    i = (i >= n) ? (2 * n - 2 - i) : i;
    return i;
}

__device__ __forceinline__ void async_copy_b128(uint32_t lds_off, uint64_t gaddr) {
    asm volatile("global_load_async_to_lds_b128 %0, %1, off"
                 :: "v"(lds_off), "v"(gaddr) : "memory");
}
__device__ __forceinline__ void async_copy_b32(uint32_t lds_off, uint64_t gaddr) {
    asm volatile("global_load_async_to_lds_b32 %0, %1, off"
                 :: "v"(lds_off), "v"(gaddr) : "memory");
}

__global__ __launch_bounds__(NTHREADS)
void median3x3_kernel(const float* __restrict__ img, float* __restrict__ out) {
    __shared__ __align__(16) float tile[LDS_ROWS * LDS_STRIDE];

    const int tid    = threadIdx.x;
    const int tileX  = blockIdx.x * TILE_W;
    const int tileY  = blockIdx.y * TILE_H;
    const int plane  = blockIdx.z;
    const bool top   = (blockIdx.y == 0);
    const bool bot   = (blockIdx.y == (IMG_H / TILE_H) - 1);
    const float* __restrict__ src = img + (size_t)plane * (IMG_H * IMG_W);
    float*       __restrict__ dst = out + (size_t)plane * (IMG_H * IMG_W);

    const uint32_t lds_base = (uint32_t)(uintptr_t)&tile[0];

#if __has_builtin(__builtin_amdgcn_tensor_load_to_lds)
    // ---- Interior tile via Tensor Data Mover: one DMA per wave (TENSORcnt).
    {
        const int w  = tid >> 5;                 // wave id 0..7
        int s0  = w * 9;                         // lds row slice start
        int cnt = LDS_ROWS - s0; cnt = cnt > 9 ? 9 : cnt;   // 9,9,...,3
        int gy0 = tileY - 1 + s0;                // global start row of slice
        if (top && s0 == 0) { s0 = 1; gy0 = 0; cnt -= 1; }  // skip reflected row -1
        if (bot && (s0 + cnt) > (LDS_ROWS - 1)) cnt -= 1;   // skip reflected row H
        const uint64_t gaddr = (uint64_t)(uintptr_t)(src + (size_t)gy0 * IMG_W + tileX);
        const uint32_t ldsa  = lds_base + (uint32_t)(s0 * LDS_STRIDE + 4) * 4u;

        u32x4 g0;
        g0[0] = 1u;                                   // count=1 valid descriptor
        g0[1] = ldsa;                                 // lds_addr (bytes)
        g0[2] = (uint32_t)gaddr;                      // global_addr[31:0]
        g0[3] = (uint32_t)(gaddr >> 32) | (2u << 30); // global_addr[56:32] | type=2

        i32x8 g1;
        g1[0] = (int)((2u << 16)       // data_size = 4 bytes
                    | (1u << 20)       // pad_enable
                    | (6u << 22)       // pad_interval: 128 DWORDs (one row)
                    | (7u << 25));     // pad_amount: 8 DWORDs -> stride 136 floats
        g1[1] = (int)(1024u << 16);    // tensor_dim0 = 1024 (low16 in [63:48])
        g1[2] = (int)(1024u << 16);    // dim0 hi=0; tensor_dim1 = 1024 (low16)
        g1[3] = (int)(128u << 16);     // dim1 hi=0; tile_dim0 = 128
        g1[4] = cnt;                   // tile_dim1 = slice rows; tile_dim2 = 0
        g1[5] = IMG_W;                 // tensor_dim0_stride = 1024 elements
        g1[6] = 0;                     // stride hi / tensor_dim1_stride lo
        g1[7] = 0;

        i32x4 g2 = {0, 0, 0, 0};       // 2-D tile: dims 2/3 unused
        i32x4 g3 = {0, 0, 0, 0};

#if __clang_major__ >= 23
        i32x8 g4 = {0, 0, 0, 0, 0, 0, 0, 0};
        __builtin_amdgcn_tensor_load_to_lds(g0, g1, g2, g3, g4, 0);
#else
        __builtin_amdgcn_tensor_load_to_lds(g0, g1, g2, g3, 0);
#endif
    }
    // Reflected border rows (only top/bottom blocks): 32 lanes x b128.
    if (top && tid < 32) {             // lds row 0 <- global row 1
        async_copy_b128(lds_base + (uint32_t)(0 * LDS_STRIDE + 4 + 4 * tid) * 4u,
                        (uint64_t)(uintptr_t)(src + (size_t)1 * IMG_W + tileX + 4 * tid));
    }
    if (bot && tid >= 32 && tid < 64) { // lds row 65 <- global row 1022
        const int t = tid - 32;
        async_copy_b128(lds_base + (uint32_t)((LDS_ROWS - 1) * LDS_STRIDE + 4 + 4 * t) * 4u,
                        (uint64_t)(uintptr_t)(src + (size_t)(IMG_H - 2) * IMG_W + tileX + 4 * t));
    }
#else
    // ---- Fallback: per-lane async b128 staging of the interior.
    {
        const int total4 = LDS_ROWS * (TILE_W / 4);   // 66 * 32 = 2112
        for (int k = tid; k < total4; k += NTHREADS) {
            const int lr  = k >> 5;
            const int lc4 = k & 31;
            const int gy  = reflect1(tileY - 1 + lr, IMG_H);
            async_copy_b128(lds_base + (uint32_t)(lr * LDS_STRIDE + 4 + 4 * lc4) * 4u,
                            (uint64_t)(uintptr_t)(src + (size_t)gy * IMG_W + tileX + 4 * lc4));
        }
    }
#endif

    // Reflected halo columns (x = tileX-1, tileX+TILE_W): per-lane b32 gathers.
    if (tid < LDS_ROWS * 2) {          // 132 lanes
        const int lr   = tid >> 1;
        const int side = tid & 1;
        const int gy   = reflect1(tileY - 1 + lr, IMG_H);
        const int gx   = reflect1(tileX - 1 + side * (TILE_W + 1), IMG_W);
        const int lcol = side ? (4 + TILE_W) : 3;
        async_copy_b32(lds_base + (uint32_t)(lr * LDS_STRIDE + lcol) * 4u,
                       (uint64_t)(uintptr_t)(src + (size_t)gy * IMG_W + gx));
    }

#if __has_builtin(__builtin_amdgcn_tensor_load_to_lds)
    __builtin_amdgcn_s_wait_tensorcnt(0);
#endif
    asm volatile("s_wait_asynccnt 0" ::: "memory");
    __syncthreads();

    // ---- Compute: thread = 4 columns x 8 rows. Row-triple (min3,med3,max3)
    // computed once per row-of-4 and reused by 3 vertically-adjacent outputs.
    const int g     = tid & 31;               // column group: cols 4g..4g+3
    const int rg    = tid >> 5;               // 0..7 (one wave per row group)
    const int rbase = rg * ROWS_PER_THREAD;
    const int lcol  = 4 + 4 * g;              // interior LDS column (16B aligned)

    float lo[3][4], mi[3][4], hi[3][4];

#define ROWREDUCE(T, SLOT)                                                    \
    {                                                                         \
        const float* row = &tile[(T) * LDS_STRIDE];                           \
        v4f   c4 = *(const v4f*)&row[lcol];     /* ds_load_b128 */            \
        float vv[6];                                                          \
        vv[0] = row[lcol - 1];                  /* ds_load_2addr pair */      \
        vv[1] = c4.x; vv[2] = c4.y; vv[3] = c4.z; vv[4] = c4.w;               \
        vv[5] = row[lcol + 4];                                                \
        _Pragma("unroll")                                                     \
        for (int i = 0; i < 4; ++i) {                                         \
            lo[SLOT][i] = fmin3(vv[i], vv[i + 1], vv[i + 2]);                 \
            mi[SLOT][i] = fmed3(vv[i], vv[i + 1], vv[i + 2]);                 \
            hi[SLOT][i] = fmax3(vv[i], vv[i + 1], vv[i + 2]);                 \
        }                                                                     \
    }

    ROWREDUCE(rbase + 0, 0);
    ROWREDUCE(rbase + 1, 1);

    float* orow = dst + (size_t)(tileY + rbase) * IMG_W + (tileX + 4 * g);
#pragma unroll
    for (int j = 0; j < ROWS_PER_THREAD; ++j) {
        const int sp = j % 3, sc = (j + 1) % 3, sn = (j + 2) % 3;
        ROWREDUCE(rbase + j + 2, sn);
        v4f o;
#pragma unroll
        for (int i = 0; i < 4; ++i) {
            o[i] = fmed3(fmax3(lo[sp][i], lo[sc][i], lo[sn][i]),
                         fmed3(mi[sp][i], mi[sc][i], mi[sn][i]),
                         fmin3(hi[sp][i], hi[sc][i], hi[sn][i]));
        }
        // Output is never re-read: non-temporal store keeps L2 for input halo reuse.
        __builtin_nontemporal_store(o, (v4f*)orow);
        orow += IMG_W;
    }
#undef ROWREDUCE
}

extern "C" void kernel_launch(void* const* d_in, const int* in_sizes, int n_in,
                              void* d_out, int out_size, void* d_ws, size_t ws_size,
                              hipStream_t stream) {
    const float* img = (const float*)d_in[0];
    float* out = (float*)d_out;
    const int planes = in_sizes[0] / (IMG_H * IMG_W);        // B*C = 24
    dim3 grid(IMG_W / TILE_W, IMG_H / TILE_H, planes);       // 8 x 16 x 24
    median3x3_kernel<<<grid, NTHREADS, 0, stream>>>(img, out);
}